// LEAP_27599459844518
// MI455X (gfx1250) — compile-verified
//
#include <hip/hip_runtime.h>
#include <hip/hip_bf16.h>
#include <math.h>

// ---------------- model constants (match reference) ----------------
#define DIMX 64
#define LTOK 48     // tokens per visit (L)
#define NVIS 32     // visits (V)
#define NB   128    // batch (B)
#define DFFX 256
#define NM1  151    // NMEDS+1
#define TST  36     // TSTEPS (fixed in harness)
#define DEFLS 0.125f  // 1/sqrt(64)

typedef __attribute__((ext_vector_type(2))) float v2f;
typedef __attribute__((ext_vector_type(8))) float v8f;
typedef unsigned int u32x4 __attribute__((ext_vector_type(4)));
typedef int i32x8 __attribute__((ext_vector_type(8)));
typedef int i32x4 __attribute__((ext_vector_type(4)));

// LDS row strides chosen as odd multiples of 4 (mod 64 banks) -> conflict-free
// A-fragment loads (16 lanes hit 16 distinct rows at the same k).
#define SX 68   // stride for 64-wide tiles
#define SW 52   // stride for 48-wide attention-weight tile

#if defined(__has_builtin)
#if __has_builtin(__builtin_amdgcn_tensor_load_to_lds) && __has_builtin(__builtin_amdgcn_s_wait_tensorcnt)
#define HAVE_TDM 1
#endif
#endif

#ifdef HAVE_TDM
// Tensor Data Mover: DMA a 2-D tile (rows x 64 fp32, row-major, contiguous)
// from global memory straight into LDS, with hardware padding of 4 DWORDs
// after every 64 DWORDs -> realizes the stride-68 bank-conflict-free layout
// without touching VGPRs. Descriptor packing per CDNA5 ISA 8.3/8.4.
// 6-arg builtin arity (clang-23 / therock-10.0 headers).
__device__ __forceinline__ void tdm_load_2d(unsigned ldsOff, const float* gsrc,
                                            int rows) {
  const unsigned rowElems = 64;
  uint64_t ga = (uint64_t)(uintptr_t)gsrc;
  u32x4 g0;
  g0.x = 0x1u;                                             // count=1, load, user
  g0.y = ldsOff;                                           // lds_addr (bytes)
  g0.z = (unsigned)ga;                                     // global_addr[31:0]
  g0.w = (unsigned)((ga >> 32) & 0x01FFFFFFu) | (2u << 30);// addr[56:32] | type=2
  i32x8 g1;
  // data_size=4B (code 2); pad_enable; pad_interval=64 DW (code 5); pad_amount=4 DW (code 3)
  g1[0] = (int)((2u << 16) | (1u << 20) | (5u << 22) | (3u << 25));
  g1[1] = (int)(rowElems << 16);                           // tensor_dim0 lo16 (bits 63:48)
  g1[2] = (int)(((unsigned)rows & 0xFFFFu) << 16);         // dim0 hi=0 | tensor_dim1 lo16
  g1[3] = (int)(rowElems << 16);                           // dim1 hi=0 | tile_dim0
  g1[4] = (int)((unsigned)rows & 0xFFFFu);                 // tile_dim1 | tile_dim2=0
  g1[5] = (int)rowElems;                                   // tensor_dim0_stride lo32
  g1[6] = 0;                                               // stride hi | dim1_stride lo
  g1[7] = 0;
  i32x4 z4 = {0, 0, 0, 0};                                 // 2-D tensor: groups 2/3 zero
  i32x8 z8 = {0, 0, 0, 0, 0, 0, 0, 0};
  __builtin_amdgcn_tensor_load_to_lds(g0, g1, z4, z4, z8, 0);
}
#endif

__device__ __forceinline__ v8f wmma4(v2f a, v2f b, v8f c) {
  // fp32 matrix path: D = A(16x4) * B(4x16) + C(16x16)
  return __builtin_amdgcn_wmma_f32_16x16x4_f32(false, a, false, b, (short)0, c, false, false);
}

// 16x16 tile at (m0,n0) of  A(row-major, stride SX) * W^T  with B[k][n] = W[n0+n][k]
__device__ __forceinline__ v8f gemm_tile_BT(const float* aS, int m0,
                                            const float* wS, int n0, int K) {
  const int lane = threadIdx.x & 31;
  const int mr = lane & 15;            // M (for A) / N (for B) within tile
  const int kh = (lane >> 4) << 1;     // 0 for lanes 0-15, 2 for lanes 16-31
  v8f acc = {};
  for (int k = 0; k < K; k += 4) {
    v2f a, b;
    a.x = aS[(m0 + mr) * SX + k + kh];
    a.y = aS[(m0 + mr) * SX + k + kh + 1];
    b.x = wS[(n0 + mr) * SX + k + kh];      // W[n][k]  == B[k][n]
    b.y = wS[(n0 + mr) * SX + k + kh + 1];
    acc = wmma4(a, b, acc);
  }
  return acc;
}

// 16x16 tile at (m0,n0) of  A(stride SW) * B(stride SX, direct B[k][n])
__device__ __forceinline__ v8f gemm_tile_B(const float* aS, int m0,
                                           const float* bS, int n0, int K) {
  const int lane = threadIdx.x & 31;
  const int mr = lane & 15;
  const int kh = (lane >> 4) << 1;
  v8f acc = {};
  for (int k = 0; k < K; k += 4) {
    v2f a, b;
    a.x = aS[(m0 + mr) * SW + k + kh];
    a.y = aS[(m0 + mr) * SW + k + kh + 1];
    b.x = bS[(k + kh) * SX + n0 + mr];
    b.y = bS[(k + kh + 1) * SX + n0 + mr];
    acc = wmma4(a, b, acc);
  }
  return acc;
}

// ---------------- generic gated attention layer --------------------
// out[b] = LayerNorm( x[b] + relu(tanh(qr_i + kr_j)) * mask_j  @  V ) with
//   q = tanh(x Wq^T + bq), k = tanh(c Wk^T), v = (c Wv^T + bv) * DEFL
// One block per batch element; 128 threads = 4 waves; fp32 WMMA throughout.
// All LDS staging is done by the Tensor Data Mover (wave 0 issues, waits on
// TENSORcnt, then a workgroup barrier publishes the tiles).
__global__ __launch_bounds__(128)
void gattn(const float* __restrict__ x, int xsb,
           const float* __restrict__ c, int csb,
           float* __restrict__ out, int osb,
           const float* __restrict__ mask, int msb,
           int Lq, int Lk,
           const float* __restrict__ Wq, const float* __restrict__ bq,
           const float* __restrict__ Wk, const float* __restrict__ wr,
           const float* __restrict__ Wv, const float* __restrict__ bv,
           const float* __restrict__ g,  const float* __restrict__ bb) {
  __shared__ float sm[3 * 48 * SX + 64 * SX + 5 * 64 + 96]; // ~57KB
  float* XS  = sm;                 // 48 x SX  (x tile, later x + attn out)
  float* CS  = XS + 48 * SX;       // 48 x SX  (c tile)
  float* T1  = CS + 48 * SX;       // 48 x SX  (q -> k -> v, time-multiplexed)
  float* WB  = T1 + 48 * SX;       // 64 x SX  (weights; reused for attn W @ stride SW)
  float* bqS = WB + 64 * SX;
  float* bvS = bqS + 64;
  float* wrS = bvS + 64;
  float* gS  = wrS + 64;
  float* bbS = gS + 64;
  float* qr  = bbS + 64;           // 48
  float* kr  = qr + 48;            // 48

  const int tid  = threadIdx.x;
  const int wv   = tid >> 5;
  const int bidx = blockIdx.x;
  const int Lqp  = (Lq + 15) & ~15;
  const int Lkp  = (Lk + 15) & ~15;

  const float* xb = x + (size_t)bidx * xsb;
  const float* cb = c + (size_t)bidx * csb;

#ifdef HAVE_TDM
  const unsigned ldsXS = (unsigned)(uintptr_t)(void*)XS;
  const unsigned ldsCS = (unsigned)(uintptr_t)(void*)CS;
  const unsigned ldsWB = (unsigned)(uintptr_t)(void*)WB;
  if (wv == 0) {
    tdm_load_2d(ldsXS, xb, Lq);
    tdm_load_2d(ldsCS, cb, Lk);
    tdm_load_2d(ldsWB, Wq, 64);
    __builtin_amdgcn_s_wait_tensorcnt(0);
  }
  // zero-fill the 16-alignment pad rows (disjoint from TDM target rows)
  for (int i = tid; i < (Lqp - Lq) * 64; i += 128)
    XS[(Lq + (i >> 6)) * SX + (i & 63)] = 0.f;
  for (int i = tid; i < (Lkp - Lk) * 64; i += 128)
    CS[(Lk + (i >> 6)) * SX + (i & 63)] = 0.f;
#else
  for (int i = tid; i < Lqp * 64; i += 128) {
    int r = i >> 6, cc = i & 63;
    XS[r * SX + cc] = (r < Lq) ? xb[r * 64 + cc] : 0.f;
  }
  for (int i = tid; i < Lkp * 64; i += 128) {
    int r = i >> 6, cc = i & 63;
    CS[r * SX + cc] = (r < Lk) ? cb[r * 64 + cc] : 0.f;
  }
  for (int i = tid; i < 4096; i += 128) WB[(i >> 6) * SX + (i & 63)] = Wq[i];
#endif
  if (tid < 64) {
    bqS[tid] = bq[tid]; bvS[tid] = bv[tid]; wrS[tid] = wr[tid];
    gS[tid]  = g[tid];  bbS[tid] = bb[tid];
  }
  __syncthreads();

  const int nqt = (Lqp >> 4) << 2;  // output tiles for Lq x 64
  const int nkt = (Lkp >> 4) << 2;  // output tiles for Lk x 64
  const int lane = tid & 31, nn = lane & 15, mo = (lane >> 4) << 3;

  // ---- Q = tanh(X Wq^T + bq) -> T1
  for (int t = wv; t < nqt; t += 4) {
    int m0 = (t >> 2) << 4, n0 = (t & 3) << 4;
    v8f d = gemm_tile_BT(XS, m0, WB, n0, 64);
    for (int r = 0; r < 8; r++) {
      int row = m0 + mo + r, col = n0 + nn;
      T1[row * SX + col] = tanhf(d[r] + bqS[col]);
    }
  }
  __syncthreads();
  for (int i = tid; i < Lq; i += 128) {
    float s = 0.f;
    for (int k = 0; k < 64; k++) s += T1[i * SX + k] * wrS[k];
    qr[i] = s;
  }
  __syncthreads();

  // ---- K = tanh(C Wk^T) -> T1   (restage Wk over Wq via TDM)
#ifdef HAVE_TDM
  if (wv == 0) { tdm_load_2d(ldsWB, Wk, 64); __builtin_amdgcn_s_wait_tensorcnt(0); }
#else
  for (int i = tid; i < 4096; i += 128) WB[(i >> 6) * SX + (i & 63)] = Wk[i];
#endif
  __syncthreads();
  for (int t = wv; t < nkt; t += 4) {
    int m0 = (t >> 2) << 4, n0 = (t & 3) << 4;
    v8f d = gemm_tile_BT(CS, m0, WB, n0, 64);
    for (int r = 0; r < 8; r++) {
      int row = m0 + mo + r, col = n0 + nn;
      T1[row * SX + col] = tanhf(d[r]);
    }
  }
  __syncthreads();
  for (int i = tid; i < Lk; i += 128) {
    float s = 0.f;
    for (int k = 0; k < 64; k++) s += T1[i * SX + k] * wrS[k];
    kr[i] = s;
  }
  __syncthreads();

  // ---- V = (C Wv^T + bv) * DEFL -> T1   (restage Wv via TDM)
#ifdef HAVE_TDM
  if (wv == 0) { tdm_load_2d(ldsWB, Wv, 64); __builtin_amdgcn_s_wait_tensorcnt(0); }
#else
  for (int i = tid; i < 4096; i += 128) WB[(i >> 6) * SX + (i & 63)] = Wv[i];
#endif
  __syncthreads();
  for (int t = wv; t < nkt; t += 4) {
    int m0 = (t >> 2) << 4, n0 = (t & 3) << 4;
    v8f d = gemm_tile_BT(CS, m0, WB, n0, 64);
    for (int r = 0; r < 8; r++) {
      int row = m0 + mo + r, col = n0 + nn;
      T1[row * SX + col] = (d[r] + bvS[col]) * DEFLS;
    }
  }
  __syncthreads();

  // ---- attention weights W[i][j] = relu(tanh(qr_i + kr_j)) * mask_j -> WB (stride SW)
  for (int i = tid; i < Lqp * Lkp; i += 128) {
    int r = i / Lkp, cc = i - r * Lkp;
    float v = 0.f;
    if (r < Lq && cc < Lk) {
      float tv = tanhf(qr[r] + kr[cc]);
      v = tv > 0.f ? tv : 0.f;
      if (mask) v *= mask[(size_t)bidx * msb + cc];
    }
    WB[r * SW + cc] = v;
  }
  __syncthreads();

  // ---- out = X + W @ V  (accumulate into XS)
  for (int t = wv; t < nqt; t += 4) {
    int m0 = (t >> 2) << 4, n0 = (t & 3) << 4;
    v8f d = gemm_tile_B(WB, m0, T1, n0, Lkp);
    for (int r = 0; r < 8; r++) {
      int row = m0 + mo + r, col = n0 + nn;
      XS[row * SX + col] += d[r];
    }
  }
  __syncthreads();

  // ---- LayerNorm + store
  float* ob = out + (size_t)bidx * osb;
  for (int i = tid; i < Lq; i += 128) {
    float mu = 0.f;
    for (int k = 0; k < 64; k++) mu += XS[i * SX + k];
    mu *= (1.f / 64.f);
    float var = 0.f;
    for (int k = 0; k < 64; k++) { float dv = XS[i * SX + k] - mu; var += dv * dv; }
    var *= (1.f / 64.f);
    float inv = rsqrtf(var + 1e-5f);
    for (int k = 0; k < 64; k++)
      ob[i * 64 + k] = (XS[i * SX + k] - mu) * inv * gS[k] + bbS[k];
  }
}

// ---------------- embedding gather ---------------------------------
__global__ __launch_bounds__(256)
void gatherk(const int* __restrict__ tok, const float* __restrict__ emb,
             float* __restrict__ E, int ntok) {
  int idx = blockIdx.x * 256 + threadIdx.x;
  if (idx < ntok * 64) {
    int t = tok[idx >> 6];
    E[idx] = emb[(size_t)t * 64 + (idx & 63)];
  }
}

// ---------------- masked max-pool over tokens + visit mask ---------
__global__ __launch_bounds__(64)
void poolk(const float* __restrict__ E, const float* __restrict__ mask,
           float* __restrict__ EV, float* __restrict__ VM) {
  __shared__ float vmsh;
  int blk = blockIdx.x, tid = threadIdx.x;
  if (tid == 0) {
    float s = 0.f;
    for (int l = 0; l < LTOK; l++) s += mask[(size_t)blk * LTOK + l];
    vmsh = (s > 0.f) ? 1.f : 0.f;
    VM[blk] = vmsh;
  }
  __syncthreads();
  float m = -3.0e38f;
  for (int l = 0; l < LTOK; l++) {
    float v = E[((size_t)blk * LTOK + l) * 64 + tid]
            + (mask[(size_t)blk * LTOK + l] - 1.f) * 1e9f;
    m = fmaxf(m, v);
  }
  EV[(size_t)blk * 64 + tid] = m * vmsh;
}

// ---------------- hsr = concat(diag, proc); mv0 = me[NMEDS+1] ------
__global__ __launch_bounds__(128)
void assemblek(const float* __restrict__ Qd, const float* __restrict__ Qp,
               const float* __restrict__ me, float* __restrict__ H,
               float* __restrict__ MV) {
  int b = blockIdx.x, tid = threadIdx.x;
  if (tid < 64) {
    H[b * 128 + tid] = Qd[b * 64 + tid];
    MV[b * 64 + tid] = me[NM1 * 64 + tid];  // me[NMEDS+1]
  } else {
    H[b * 128 + tid] = Qp[b * 64 + (tid - 64)];
  }
}

// ---------------- per-step MLP head + softmax feedback -------------
__global__ __launch_bounds__(256)
void ffnk(const float* __restrict__ H, const float* __restrict__ Wdff,
          const float* __restrict__ bdff, const float* __restrict__ Wfin,
          const float* __restrict__ bfin, const float* __restrict__ me,
          float* __restrict__ MV, float* __restrict__ out, int t) {
  __shared__ float hS[128];
  __shared__ float xS[DFFX];
  __shared__ float pS[NM1];
  __shared__ float invS;
  int b = blockIdx.x, tid = threadIdx.x;
  if (tid < 128) hS[tid] = H[b * 128 + tid];
  __syncthreads();
  {
    // x = sum over the 2 rows of (h Wdff^T + bdff), then relu
    float acc = 2.f * bdff[tid];
    const float* w = Wdff + (size_t)tid * 64;
    for (int k = 0; k < 64; k++) acc += (hS[k] + hS[64 + k]) * w[k];
    xS[tid] = acc > 0.f ? acc : 0.f;
  }
  __syncthreads();
  if (tid < NM1) {
    float p = bfin[tid];
    const float* w = Wfin + (size_t)tid * DFFX;
    for (int f = 0; f < DFFX; f++) p += xS[f] * w[f];
    pS[tid] = p;
    out[((size_t)b * TST + t) * NM1 + tid] = 1.f / (1.f + expf(-p));
  }
  __syncthreads();
  if (tid == 0) {
    float mx = -3.0e38f;
    for (int j = 0; j < NM1; j++) mx = fmaxf(mx, pS[j]);
    float s = 0.f;
    for (int j = 0; j < NM1; j++) { float e = expf(pS[j] - mx); pS[j] = e; s += e; }
    invS = 1.f / s;
  }
  __syncthreads();
  if (tid < 64) {
    float acc = 0.f;
    for (int j = 0; j < NM1; j++) acc += pS[j] * me[(size_t)j * 64 + tid];
    MV[b * 64 + tid] = acc * invS;
  }
}

// ---------------- host orchestration -------------------------------
extern "C" void kernel_launch(void* const* d_in, const int* in_sizes, int n_in,
                              void* d_out, int out_size, void* d_ws, size_t ws_size,
                              hipStream_t stream) {
  (void)in_sizes; (void)n_in; (void)out_size; (void)ws_size;
  // setup_inputs() insertion order, params dicts flattened recursively:
  // 0:p 1:d 2:pm 3:dm 4:T 5:diag_embed 6:proc_embed 7:med_embed
  // 8..15 diag_med{Wq,bq,Wk,wr,Wv,bv,g,b} 16..23 diag_visit 24..31 proc_med
  // 32..39 proc_visit 40..47 leap_med 48:Wdff 49:bdff 50:Wfin 51:bfin
  const int*   p_tok = (const int*)d_in[0];
  const int*   d_tok = (const int*)d_in[1];
  const float* pm    = (const float*)d_in[2];
  const float* dm    = (const float*)d_in[3];
  const float* diag_embed = (const float*)d_in[5];
  const float* proc_embed = (const float*)d_in[6];
  const float* med_embed  = (const float*)d_in[7];
  const float* DMp[8]; const float* DVp[8]; const float* PMp[8];
  const float* PVp[8]; const float* LMp[8];
  for (int i = 0; i < 8; i++) {
    DMp[i] = (const float*)d_in[8 + i];
    DVp[i] = (const float*)d_in[16 + i];
    PMp[i] = (const float*)d_in[24 + i];
    PVp[i] = (const float*)d_in[32 + i];
    LMp[i] = (const float*)d_in[40 + i];
  }
  const float* Wdff = (const float*)d_in[48];
  const float* bdff = (const float*)d_in[49];
  const float* Wfin = (const float*)d_in[50];
  const float* bfin = (const float*)d_in[51];

  // workspace layout (floats), ~52 MB total; E reused across the two streams
  float* E  = (float*)d_ws;                          // B*V*L*64
  float* EV = E  + (size_t)NB * NVIS * LTOK * 64;    // B*V*64
  float* VM = EV + (size_t)NB * NVIS * 64;           // B*V
  float* Qd = VM + (size_t)NB * NVIS;                // B*64
  float* Qp = Qd + (size_t)NB * 64;                  // B*64
  float* H  = Qp + (size_t)NB * 64;                  // B*2*64
  float* MV = H  + (size_t)NB * 128;                 // B*64

  auto launch_ga = [&](const float* x, int xsb, const float* c, int csb,
                       float* outp, int osb, const float* mask, int msb,
                       int Lq, int Lk, const float* const* P, int l, int nblk) {
    gattn<<<nblk, 128, 0, stream>>>(
        x, xsb, c, csb, outp, osb, mask, msb, Lq, Lk,
        P[0] + (size_t)l * 4096, P[1] + (size_t)l * 64,
        P[2] + (size_t)l * 4096, P[3] + (size_t)l * 64,
        P[4] + (size_t)l * 4096, P[5] + (size_t)l * 64,
        P[6] + (size_t)l * 64,   P[7] + (size_t)l * 64);
  };

  for (int s = 0; s < 2; s++) {  // s=0: diag, s=1: proc
    const int*   tok = (s == 0) ? d_tok : p_tok;
    const float* msk = (s == 0) ? dm : pm;
    const float* emb = (s == 0) ? diag_embed : proc_embed;
    const float* const* med = (s == 0) ? DMp : PMp;
    const float* const* vis = (s == 0) ? DVp : PVp;
    float* Q = (s == 0) ? Qd : Qp;

    const int ntok = NB * NVIS * LTOK;
    gatherk<<<(ntok * 64 + 255) / 256, 256, 0, stream>>>(tok, emb, E, ntok);

    // med-level self-attention: 4096 batches of Lq=Lk=48 (WMMA heavy)
    for (int l = 0; l < 4; l++)
      launch_ga(E, LTOK * 64, E, LTOK * 64, E, LTOK * 64,
                msk, LTOK, LTOK, LTOK, med, l, NB * NVIS);

    poolk<<<NB * NVIS, 64, 0, stream>>>(E, msk, EV, VM);

    // visit-level: q = last visit row, kv = first V-1 rows, keys masked by vm
    launch_ga(EV + (NVIS - 1) * 64, NVIS * 64, EV, NVIS * 64,
              Q, 64, VM, NVIS, 1, NVIS - 1, vis, 0, NB);
    for (int l = 1; l < 4; l++)
      launch_ga(Q, 64, EV, NVIS * 64, Q, 64, VM, NVIS, 1, NVIS - 1, vis, l, NB);
  }

  assemblek<<<NB, 128, 0, stream>>>(Qd, Qp, med_embed, H, MV);

  for (int t = 0; t < TST; t++) {
    for (int l = 0; l < 4; l++)
      launch_ga(H, 128, MV, 64, H, 128, nullptr, 0, 2, 1, LMp, l, NB);
    ffnk<<<NB, 256, 0, stream>>>(H, Wdff, bdff, Wfin, bfin, med_embed,
                                 MV, (float*)d_out, t);
  }
}